// patches_generator_12043088298456
// MI455X (gfx1250) — compile-verified
//
#include <hip/hip_runtime.h>

// Problem constants (match the reference).
#define N_      16
#define C_      128
#define H_      112
#define W_      112
#define P_      98
#define S_      256                 // 16x16 samples per patch
#define HW_     (H_ * W_)           // 12544 floats per channel plane
#define PLANE_B (HW_ * 4)           // 50176 bytes per plane
#define NPS_    (P_ * S_)           // 25088 samples per image
#define NSAMP_  (N_ * NPS_)         // 401408 samples total
#define CG_     4                   // channels staged per block (contiguous)
#define LDSB_   (CG_ * PLANE_B)     // 200704 B of LDS (<= 320KB WGP LDS)

// 8-byte vector with 4-byte alignment (neighbor columns are dword-aligned).
typedef float v2f __attribute__((ext_vector_type(2), aligned(4)));

// ---------------------------------------------------------------------------
// Kernel 1: per-sample setup, computed ONCE for the whole problem.
//   ws layout: float4 w[NSAMP_]  (w00,w01,w10,w11)   at ws + 0
//              int    off[NSAMP_] (byte offset in plane) at ws + NSAMP_*16
// ---------------------------------------------------------------------------
__global__ __launch_bounds__(256)
void setup_kernel(const float* __restrict__ grid,
                  float4* __restrict__ wts,
                  int* __restrict__ offs)
{
    const int g = blockIdx.x * 256 + threadIdx.x;      // < NSAMP_ (exact grid)

    const v2f gv = __builtin_nontemporal_load((const v2f*)(grid + (size_t)g * 2));

    // unnormalize (align_corners=False) + border clamp
    float x = fminf(fmaxf((gv.x + 1.0f) * (0.5f * W_) - 0.5f, 0.0f), (float)(W_ - 1));
    float y = fminf(fmaxf((gv.y + 1.0f) * (0.5f * H_) - 0.5f, 0.0f), (float)(H_ - 1));
    float xf = floorf(x), yf = floorf(y);
    float wx = x - xf,    wy = y - yf;
    int   x0 = (int)xf,   y0 = (int)yf;

    // Fold the right/bottom border clamp into the weights so the 4 taps are
    // always cols (xa, xa+1) x rows (ya, ya+1) with xa<=W-2, ya<=H-2.
    // At the clamp the fractional weight is exactly 0, so (0,1) is exact.
    const bool cx = (x0 >= W_ - 1);
    const bool cy = (y0 >= H_ - 1);
    const int  xa = cx ? (W_ - 2) : x0;
    const int  ya = cy ? (H_ - 2) : y0;
    const float ax = cx ? 0.0f : (1.0f - wx);
    const float bx = cx ? 1.0f : wx;
    const float c0 = cy ? 0.0f : (1.0f - wy);
    const float c1 = cy ? 1.0f : wy;

    float4 w;
    w.x = c0 * ax;  w.y = c0 * bx;      // row ya  : cols xa, xa+1
    w.z = c1 * ax;  w.w = c1 * bx;      // row ya+1: cols xa, xa+1

    wts[g]  = w;
    offs[g] = (ya * W_ + xa) << 2;      // byte offset within one channel plane
}

// ---------------------------------------------------------------------------
// Kernel 2: one block per (image n, 4-channel group). Stage the 4 contiguous
// channel planes (200704 B) into LDS via gfx1250 async-copy-to-LDS, then do
// all 25088 bilinear gathers for this image from LDS (ds_read2-friendly
// offsets 0/1 and 112/113 dwords). Every plane is read from global exactly
// once kernel-wide; all global traffic is coalesced streams.
// ---------------------------------------------------------------------------
__global__ __launch_bounds__(512)
void sample_main_kernel(const float* __restrict__ fm,
                        const float4* __restrict__ wts,
                        const int* __restrict__ offs,
                        float* __restrict__ out)
{
    __shared__ float lds[CG_ * HW_];                 // 200704 B static LDS

    const int n  = blockIdx.x >> 5;                  // 16 images
    const int c0 = (blockIdx.x & 31) * CG_;          // 32 groups of 4 channels

    // ---- stage 4 contiguous planes with async global->LDS b128 copies ----
    const char* src   = (const char*)(fm + ((size_t)n * C_ + c0) * HW_);
    const unsigned lb = (unsigned)(uintptr_t)(void*)lds;   // LDS byte address
    const unsigned t16 = threadIdx.x * 16u;
#pragma unroll
    for (int k = 0; k < 25; ++k) {                   // 25 * 512 * 16B covers 200704B
        unsigned ofs = (unsigned)k * 8192u + t16;
        if (ofs < LDSB_) {
            unsigned           laddr = lb + ofs;
            unsigned long long gaddr = (unsigned long long)(src + ofs);
            asm volatile("global_load_async_to_lds_b128 %0, %1, off"
                         :: "v"(laddr), "v"(gaddr) : "memory");
        }
    }
    asm volatile("s_wait_asynccnt 0x0" ::: "memory");
    __syncthreads();

    // ---- bilinear sampling for all samples of image n, 4 channels each ----
    const float4* wt = wts  + (size_t)n * NPS_;
    const int*    of = offs + (size_t)n * NPS_;
    float* outn = out + (size_t)n * P_ * C_ * S_;

#pragma unroll 2
    for (int i = threadIdx.x; i < NPS_; i += 512) {
        const int    word = of[i] >> 2;              // dword offset in plane
        const float4 w    = wt[i];
        const int p = i >> 8;                        // patch within image
        const int s = i & 255;                       // sample within patch
        float* po = outn + ((size_t)p * C_ + c0) * S_ + s;

#pragma unroll
        for (int j = 0; j < CG_; ++j) {
            const int b = j * HW_ + word;
            float v =      w.x * lds[b];             // (ya  , xa  )
            v = fmaf(w.y, lds[b + 1],      v);       // (ya  , xa+1)
            v = fmaf(w.z, lds[b + W_],     v);       // (ya+1, xa  )
            v = fmaf(w.w, lds[b + W_ + 1], v);       // (ya+1, xa+1)
            // 205MB write-once output: stream past L2 (keep planes resident)
            __builtin_nontemporal_store(v, po + (size_t)j * S_);
        }
    }
}

// ---------------------------------------------------------------------------
// Fallback (round-0 kernel): direct L2 gathers. Used only if the
// workspace is too small for the setup table.
// ---------------------------------------------------------------------------
__global__ __launch_bounds__(256)
void grid_sample_fallback_kernel(const float* __restrict__ fm,
                                 const float* __restrict__ grid,
                                 float* __restrict__ out)
{
    const int s   = threadIdx.x;
    const int bid = blockIdx.x;                      // n*P_ + p
    const int n   = bid / P_;

    const v2f g = __builtin_nontemporal_load(
        (const v2f*)(grid + ((size_t)bid * S_ + s) * 2));

    float x = fminf(fmaxf((g.x + 1.0f) * (0.5f * W_) - 0.5f, 0.0f), (float)(W_ - 1));
    float y = fminf(fmaxf((g.y + 1.0f) * (0.5f * H_) - 0.5f, 0.0f), (float)(H_ - 1));
    float xf = floorf(x), yf = floorf(y);
    float wx = x - xf,    wy = y - yf;
    int   x0 = (int)xf,   y0 = (int)yf;

    const bool cx = (x0 >= W_ - 1);
    const bool cy = (y0 >= H_ - 1);
    const int  xa = cx ? (W_ - 2) : x0;
    const int  ya = cy ? (H_ - 2) : y0;
    const float ax = cx ? 0.0f : (1.0f - wx);
    const float bx = cx ? 1.0f : wx;
    const float c0 = cy ? 0.0f : (1.0f - wy);
    const float c1 = cy ? 1.0f : wy;
    const float w00 = c0 * ax, w01 = c0 * bx, w10 = c1 * ax, w11 = c1 * bx;

    const size_t o0 = (size_t)ya * W_ + xa;
    const float* __restrict__ fmn  = fm  + (size_t)n   * C_ * HW_;
    float*       __restrict__ outp = out + (size_t)bid * C_ * S_ + s;

#pragma unroll 4
    for (int c = 0; c < C_; ++c) {
        const float* base = fmn + (size_t)c * HW_;
        v2f r0 = *(const v2f*)(base + o0);
        v2f r1 = *(const v2f*)(base + o0 + W_);
        float v =      w00 * r0.x;
        v = fmaf(w01, r0.y, v);
        v = fmaf(w10, r1.x, v);
        v = fmaf(w11, r1.y, v);
        __builtin_nontemporal_store(v, outp + (size_t)c * S_);
    }
}

extern "C" void kernel_launch(void* const* d_in, const int* in_sizes, int n_in,
                              void* d_out, int out_size, void* d_ws, size_t ws_size,
                              hipStream_t stream)
{
    (void)in_sizes; (void)n_in; (void)out_size;
    const float* fm   = (const float*)d_in[0];   // [N, C, H, W]
    const float* grid = (const float*)d_in[1];   // [N, P, 16, 16, 2]
    float*       out  = (float*)d_out;           // [N, P, C, 16, 16]

    const size_t ws_needed = (size_t)NSAMP_ * 16 + (size_t)NSAMP_ * 4;  // ~8MB

    if (ws_size >= ws_needed) {
        float4* wts = (float4*)d_ws;
        int*    off = (int*)((char*)d_ws + (size_t)NSAMP_ * 16);
        // 1) per-sample weights/offsets, once
        setup_kernel<<<NSAMP_ / 256, 256, 0, stream>>>(grid, wts, off);
        // 2) LDS-staged bilinear sampling: 16 images x 32 channel-groups
        sample_main_kernel<<<N_ * 32, 512, 0, stream>>>(fm, wts, off, out);
    } else {
        grid_sample_fallback_kernel<<<N_ * P_, 256, 0, stream>>>(fm, grid, out);
    }
}